// GResBlock_46780783788357
// MI455X (gfx1250) — compile-verified
//
#include <hip/hip_runtime.h>

typedef __attribute__((ext_vector_type(2))) float v2f;
typedef __attribute__((ext_vector_type(8))) float v8f;

// Guaranteed-native f32 atomic add (no return): lowers to the CDNA5
// GLOBAL_ATOMIC_ADD_F32 instruction (STOREcnt-tracked; s_endpgm drains it).
// Avoids any chance of a CAS-loop fallback on the 410M-atomic hot path.
__device__ __forceinline__ void gatomic_fadd(float* p, float v) {
    asm volatile("global_atomic_add_f32 %0, %1, off"
                 :
                 : "v"(p), "v"(v)
                 : "memory");
}

// ---------------- degree / normalization ----------------

__global__ void k_fill1(float* __restrict__ deg, int n) {
    int i = blockIdx.x * blockDim.x + threadIdx.x;
    if (i < n) deg[i] = 1.0f;  // self-loop contributes 1 to every degree
}

__global__ void k_deg(const int* __restrict__ dst, float* __restrict__ deg, int e) {
    int i = blockIdx.x * blockDim.x + threadIdx.x;
    if (i < e) gatomic_fadd(&deg[dst[i]], 1.0f);
}

__global__ void k_rsqrt(float* __restrict__ d, int n) {
    int i = blockIdx.x * blockDim.x + threadIdx.x;
    if (i < n) d[i] = rsqrtf(d[i]);  // deg >= 1 always (self loops)
}

// ---------------- f32 WMMA GEMM: H[N,128] = X[N,128] @ W[128,128] ----------------
// One 256-thread block = 8 waves; wave w computes the 16x16 tile at
// (rows blockIdx.x*16, cols w*16). K=128 swept in 32 steps of V_WMMA_F32_16X16X4_F32.
//
// ISA 32-bit layouts (CDNA5 ISA §7.12.2):
//   A 16x4 : lane&15 = M ; VGPR v holds K = 2*(lane>>4) + v
//   B 4x16 : lane&15 = N ; VGPR v holds K = 2*(lane>>4) + v
//   C/D    : lane&15 = N ; VGPR v holds M = v + 8*(lane>>4)

__global__ void k_gemm16(const float* __restrict__ X, const float* __restrict__ W,
                         float* __restrict__ H, int n) {
    const int wave = threadIdx.x >> 5;
    const int lane = threadIdx.x & 31;
    const int l15  = lane & 15;
    const int hi   = lane >> 4;
    const int m0   = blockIdx.x * 16;
    const int n0   = wave * 16;

    int row = m0 + l15;
    if (row >= n) row = n - 1;            // EXEC must stay all-ones for WMMA
    const float* xr = X + (size_t)row * 128;

    v8f c = {};
#pragma unroll
    for (int k0 = 0; k0 < 128; k0 += 4) {
        const int k = k0 + 2 * hi;        // this lane's K pair {k, k+1}
        v2f a = *(const v2f*)(xr + k);    // 8B-aligned (k even, row 512B-aligned)
        v2f b;
        b.x = W[(size_t)k       * 128 + n0 + l15];
        b.y = W[(size_t)(k + 1) * 128 + n0 + l15];
        c = __builtin_amdgcn_wmma_f32_16x16x4_f32(
                /*neg_a=*/false, a, /*neg_b=*/false, b,
                /*c_mod=*/(short)0, c, /*reuse_a=*/false, /*reuse_b=*/false);
    }

#pragma unroll
    for (int v = 0; v < 8; ++v) {
        const int r = m0 + v + 8 * hi;
        if (r < n) H[(size_t)r * 128 + (n0 + l15)] = c[v];
    }
}

// ---------------- aggregation init: acc = bias + h * dinv^2  (self loop) ----------------

__global__ void k_agg_init(const float* __restrict__ H, const float* __restrict__ dinv,
                           const float* __restrict__ bias, float* __restrict__ acc,
                           long long total) {
    long long i = (long long)blockIdx.x * blockDim.x + threadIdx.x;
    if (i >= total) return;
    const int  r  = (int)(i >> 7);
    const int  ch = (int)(i & 127);
    const float di = dinv[r];
    acc[i] = bias[ch] + H[i] * di * di;
}

// ---------------- edge scatter: acc[dst] += h[src] * dinv[src]*dinv[dst] ----------------
// 32 lanes per edge, float4 per lane => one coalesced 512B gather burst per edge
// (L2-resident since h fits in the 192MB L2), then 4 native f32 atomics per lane
// into the L2-resident accumulator.

__global__ void k_scatter(const float* __restrict__ H, const float* __restrict__ dinv,
                          const int* __restrict__ src, const int* __restrict__ dst,
                          float* __restrict__ acc, int e) {
    long long t    = (long long)blockIdx.x * blockDim.x + threadIdx.x;
    long long edge = t >> 5;
    if (edge >= e) return;
    const int chunk = (int)(t & 31);

    const int s = src[edge];
    const int d = dst[edge];
    const float nrm = dinv[s] * dinv[d];

    const float4 v = ((const float4*)(H + (size_t)s * 128))[chunk];
    float* o = acc + (size_t)d * 128 + (size_t)chunk * 4;
    gatomic_fadd(o + 0, v.x * nrm);
    gatomic_fadd(o + 1, v.y * nrm);
    gatomic_fadd(o + 2, v.z * nrm);
    gatomic_fadd(o + 3, v.w * nrm);
}

// ---------------- elementwise ----------------

__global__ void k_relu(const float* __restrict__ in, float* __restrict__ out, long long total) {
    long long i = (long long)blockIdx.x * blockDim.x + threadIdx.x;
    if (i < total) out[i] = fmaxf(in[i], 0.0f);
}

// out = (x + relu(out)) * 0.5 ; x streamed with non-temporal hint so the 51MB
// stream does not evict the L2-resident feature/accumulator working set.
__global__ void k_final(const float* __restrict__ x, float* __restrict__ out, long long total) {
    long long i = (long long)blockIdx.x * blockDim.x + threadIdx.x;
    if (i < total) {
        const float xv = __builtin_nontemporal_load(x + i);
        out[i] = (xv + fmaxf(out[i], 0.0f)) * 0.5f;
    }
}

// ---------------- driver ----------------

extern "C" void kernel_launch(void* const* d_in, const int* in_sizes, int n_in,
                              void* d_out, int out_size, void* d_ws, size_t ws_size,
                              hipStream_t stream) {
    const float* x   = (const float*)d_in[0];
    const int* edges = (const int*)d_in[1];
    const float* W1  = (const float*)d_in[2];
    const float* b1  = (const float*)d_in[3];
    const float* W2  = (const float*)d_in[4];
    const float* b2  = (const float*)d_in[5];

    const int N = in_sizes[0] / 128;
    const int E = in_sizes[1] / 2;
    const int* src = edges;
    const int* dst = edges + E;
    float* out = (float*)d_out;

    char* ws = (char*)d_ws;
    float* dinv = (float*)ws;
    size_t off = ((size_t)N * sizeof(float) + 255) & ~(size_t)255;
    float* buf1 = (float*)(ws + off);                 // N*128 floats
    float* buf2 = buf1 + (size_t)N * 128;             // N*128 floats

    const int T = 256;
    const long long total = (long long)N * 128;
    auto cdiv = [](long long a, long long b) { return (unsigned)((a + b - 1) / b); };

    // normalization coefficients (edge-structure only; shared by both layers)
    k_fill1 <<<cdiv(N, T), T, 0, stream>>>(dinv, N);
    k_deg   <<<cdiv(E, T), T, 0, stream>>>(dst, dinv, E);
    k_rsqrt <<<cdiv(N, T), T, 0, stream>>>(dinv, N);

    // ---- layer 1 ----
    k_gemm16  <<<cdiv(N, 16), 256, 0, stream>>>(x, W1, buf1, N);
    k_agg_init<<<cdiv(total, T), T, 0, stream>>>(buf1, dinv, b1, buf2, total);
    k_scatter <<<cdiv((long long)E * 32, T), T, 0, stream>>>(buf1, dinv, src, dst, buf2, E);
    k_relu    <<<cdiv(total, T), T, 0, stream>>>(buf2, buf1, total);

    // ---- layer 2 (accumulate directly into d_out) ----
    k_gemm16  <<<cdiv(N, 16), 256, 0, stream>>>(buf1, W2, buf2, N);
    k_agg_init<<<cdiv(total, T), T, 0, stream>>>(buf2, dinv, b2, out, total);
    k_scatter <<<cdiv((long long)E * 32, T), T, 0, stream>>>(buf2, dinv, src, dst, out, E);

    // residual + scale: out = (x + relu(out)) * 0.5
    k_final   <<<cdiv(total, T), T, 0, stream>>>(x, out, total);
}